// FactorizationMachine_51470888075720
// MI455X (gfx1250) — compile-verified
//
#include <hip/hip_runtime.h>

// FM: out[n] = bias + x_n.h + 0.5*( ||V x_n||^2 - sum_j w_j x_nj^2 ),  w_j = sum_k V_kj^2
// N=16384, D=4096, K=32.  Memory floor: 256MB of x @ 23.3 TB/s ~= 11.5us.
// s = x @ V^T (4.3 GFLOP f32) runs on V_WMMA_F32_16X16X4_F32; each wave owns TWO
// 16-row tiles of x so every B/h/w broadcast load feeds 4 WMMAs.

typedef __attribute__((ext_vector_type(2))) float v2f;
typedef __attribute__((ext_vector_type(8))) float v8f;

#define N_ROWS 16384
#define DIM    4096
#define KFAC   32

// ---- kernel 1: w[j] = sum_k V[k][j]^2  (V is K x D row-major; 512KB read, trivial)
__global__ __launch_bounds__(256) void fm_prep_w(const float* __restrict__ V,
                                                 float* __restrict__ w) {
    int j = blockIdx.x * blockDim.x + threadIdx.x;
    if (j < DIM) {
        float s = 0.f;
#pragma unroll
        for (int k = 0; k < KFAC; ++k) {
            float v = V[(size_t)k * DIM + j];
            s = fmaf(v, v, s);
        }
        w[j] = s;
    }
}

// ---- kernel 2: streaming FM. One wave <-> two 16-row tiles of x (32 rows).
// A-matrix f32 16x4 layout: lane L, VGPR v  ->  M = L%16, K = v + 2*(L>=16),
// so lane L's two A floats are x[m0 + L%16][j + 2*(L>=16) + {0,1}] == one b64 load.
// B (=V^T tile) has the identical lane pattern over rows of V.
// C/D f32 16x16 layout: VGPR v, lane L -> M = v + 8*(L>=16), N = L%16.
__global__ __launch_bounds__(256) void fm_main(const float* __restrict__ x,
                                               const float* __restrict__ h,
                                               const float* __restrict__ bias,
                                               const float* __restrict__ V,
                                               const float* __restrict__ w,
                                               float* __restrict__ out) {
    const int lane = threadIdx.x & 31;
    const int wave = threadIdx.x >> 5;
    const int pair = blockIdx.x * 8 + wave;   // 32-row tile pair (exactly 512)
    const int m0   = pair * 32;
    const int half = lane >> 4;               // 0: lanes 0-15, 1: lanes 16-31
    const int nidx = lane & 15;
    const int coff = 2 * half;                // column offset inside each 4-wide chunk

    const float* xp  = x + (size_t)(m0 + nidx) * DIM + coff;   // A tile 0 stream
    const float* vp0 = V + (size_t)nidx * DIM + coff;          // B, factors 0..15
    const float* vp1 = V + (size_t)(nidx + 16) * DIM + coff;   // B, factors 16..31
    const float* hp  = h + coff;
    const float* wp  = w + coff;

    v8f acc00 = {};  // tile0 x factors 0..15
    v8f acc01 = {};  // tile0 x factors 16..31
    v8f acc10 = {};  // tile1 x factors 0..15
    v8f acc11 = {};  // tile1 x factors 16..31
    v2f lin0 = {0.f, 0.f}, lin1 = {0.f, 0.f};  // per-lane x.h partials (pair = 2 cols)
    v2f wq0  = {0.f, 0.f}, wq1  = {0.f, 0.f};  // per-lane w*x^2 partials

    for (int j = 0; j < DIM; j += 4) {
        v2f a0 = *(const v2f*)(xp + j);                 // rows m0..m0+15
        v2f a1 = *(const v2f*)(xp + 16 * DIM + j);      // rows m0+16..m0+31 (imm offset)
        v2f b0 = *(const v2f*)(vp0 + j);
        v2f b1 = *(const v2f*)(vp1 + j);
        v2f hh = *(const v2f*)(hp + j);
        v2f ww = *(const v2f*)(wp + j);

        // speculative stream prefetch ~2KB ahead (silently dropped if OOB)
        __builtin_prefetch((const void*)(xp + j + 512), 0, 1);
        __builtin_prefetch((const void*)(xp + 16 * DIM + j + 512), 0, 1);

        // D = A*B + C (f32, full precision): 4 WMMAs per shared B load
        acc00 = __builtin_amdgcn_wmma_f32_16x16x4_f32(false, a0, false, b0,
                                                      (short)0, acc00, false, false);
        acc01 = __builtin_amdgcn_wmma_f32_16x16x4_f32(false, a0, false, b1,
                                                      (short)0, acc01, false, false);
        acc10 = __builtin_amdgcn_wmma_f32_16x16x4_f32(false, a1, false, b0,
                                                      (short)0, acc10, false, false);
        acc11 = __builtin_amdgcn_wmma_f32_16x16x4_f32(false, a1, false, b1,
                                                      (short)0, acc11, false, false);

        // keep accumulators as v2f so pk-FMAs form without register shuffling
        lin0 = a0 * hh + lin0;
        lin1 = a1 * hh + lin1;
        wq0  = (a0 * a0) * ww + wq0;
        wq1  = (a1 * a1) * ww + wq1;
    }

    const float b = bias[0];

    // ---- tile 0 epilogue ----
    {
        float lin = lin0.x + lin0.y;
        float wq  = wq0.x + wq0.y;
        lin += __shfl_xor(lin, 16, 32);   // lanes L, L^16 cover all 4 cols of row L%16
        wq  += __shfl_xor(wq,  16, 32);
#pragma unroll
        for (int v = 0; v < 8; ++v) {
            float t = acc00[v] * acc00[v] + acc01[v] * acc01[v];
            t += __shfl_xor(t, 1, 32);
            t += __shfl_xor(t, 2, 32);
            t += __shfl_xor(t, 4, 32);
            t += __shfl_xor(t, 8, 32);
            // lane v -> row v (half 0); lane v+24 -> row v+8 (half 1)
            if (lane == v || lane == v + 24) {
                int row = half ? (v + 8) : v;
                out[m0 + row] = b + lin + 0.5f * (t - wq);
            }
        }
    }
    // ---- tile 1 epilogue ----
    {
        float lin = lin1.x + lin1.y;
        float wq  = wq1.x + wq1.y;
        lin += __shfl_xor(lin, 16, 32);
        wq  += __shfl_xor(wq,  16, 32);
#pragma unroll
        for (int v = 0; v < 8; ++v) {
            float t = acc10[v] * acc10[v] + acc11[v] * acc11[v];
            t += __shfl_xor(t, 1, 32);
            t += __shfl_xor(t, 2, 32);
            t += __shfl_xor(t, 4, 32);
            t += __shfl_xor(t, 8, 32);
            if (lane == v || lane == v + 24) {
                int row = half ? (v + 8) : v;
                out[m0 + 16 + row] = b + lin + 0.5f * (t - wq);
            }
        }
    }
}

extern "C" void kernel_launch(void* const* d_in, const int* in_sizes, int n_in,
                              void* d_out, int out_size, void* d_ws, size_t ws_size,
                              hipStream_t stream) {
    const float* x    = (const float*)d_in[0];
    const float* h    = (const float*)d_in[1];
    const float* bias = (const float*)d_in[2];
    const float* V    = (const float*)d_in[3];
    float*       out  = (float*)d_out;
    float*       w    = (float*)d_ws;          // 4096 floats of scratch

    fm_prep_w<<<DIM / 256, 256, 0, stream>>>(V, w);
    fm_main<<<(N_ROWS / 32) / 8, 256, 0, stream>>>(x, h, bias, V, w, out);
}